// GNNModel_30313879176002
// MI455X (gfx1250) — compile-verified
//
#include <hip/hip_runtime.h>
#include <math.h>

typedef __attribute__((ext_vector_type(2))) float v2f;
typedef __attribute__((ext_vector_type(8))) float v8f;

static constexpr int Dh = 64;   // input dim == hidden dim
static constexpr int Oc = 10;   // output classes

// ---------------------------------------------------------------------------
// Utility: fill a float buffer with a constant
// ---------------------------------------------------------------------------
__global__ void fill_kernel(float* __restrict__ p, float v, int n) {
  int i = blockIdx.x * blockDim.x + threadIdx.x;
  if (i < n) p[i] = v;
}

// deg[dst] += 1 for each edge (deg buffer pre-filled with 1.0 for self-loop)
__global__ void deg_count_kernel(const int* __restrict__ dst, float* deg, int e) {
  int i = blockIdx.x * blockDim.x + threadIdx.x;
  if (i < e) unsafeAtomicAdd(&deg[dst[i]], 1.0f);
}

// in-place deg -> 1/sqrt(deg)   (deg >= 1 guaranteed)
__global__ void rsqrt_kernel(float* __restrict__ p, int n) {
  int i = blockIdx.x * blockDim.x + threadIdx.x;
  if (i < n) p[i] = rsqrtf(p[i]);
}

// ---------------------------------------------------------------------------
// GEMM: C[nrows,64] = A[nrows,64] * W[64,64]  using V_WMMA_F32_16X16X4_F32.
// Block = 128 threads (4 waves). Block handles one 16-row stripe; each wave
// owns one 16-column tile. A stripe (4KB) and W (16KB) staged in LDS.
// ---------------------------------------------------------------------------
__global__ __launch_bounds__(128) void gemm64_wmma_kernel(
    const float* __restrict__ A, const float* __restrict__ W,
    float* __restrict__ C, int nrows) {
  __shared__ float sA[16 * Dh];
  __shared__ float sW[Dh * Dh];
  const int tid  = threadIdx.x;
  const int wave = tid >> 5;        // 0..3 -> column tile
  const int lane = tid & 31;
  const int r0   = blockIdx.x * 16; // first row of stripe

  // stage A stripe: 1024 floats = 256 float4, 128 threads -> 2 iters
  {
    const float4* srcA = (const float4*)(A + (size_t)r0 * Dh);
    float4* dstA = (float4*)sA;
    #pragma unroll
    for (int i = tid; i < 16 * Dh / 4; i += 128) dstA[i] = srcA[i];
  }
  // stage W: 4096 floats = 1024 float4, 128 threads -> 8 iters
  {
    const float4* srcW = (const float4*)W;
    float4* dstW = (float4*)sW;
    #pragma unroll
    for (int i = tid; i < Dh * Dh / 4; i += 128) dstW[i] = srcW[i];
  }
  __syncthreads();

  const int m  = lane & 15;          // row (A frag) / col (B frag) within tile
  const int kh = (lane >> 4) * 2;    // K sub-offset: lanes 16-31 hold K=2,3
  const int c0 = wave * 16;          // column tile base

  v8f acc = {};
  #pragma unroll
  for (int kk = 0; kk < Dh; kk += 4) {
    v2f a, b;
    a.x = sA[m * Dh + kk + kh + 0];
    a.y = sA[m * Dh + kk + kh + 1];
    b.x = sW[(kk + kh + 0) * Dh + c0 + m];
    b.y = sW[(kk + kh + 1) * Dh + c0 + m];
    acc = __builtin_amdgcn_wmma_f32_16x16x4_f32(
        /*neg_a=*/false, a, /*neg_b=*/false, b,
        /*c_mod=*/(short)0, acc, /*reuse_a=*/false, /*reuse_b=*/false);
  }

  // C/D layout: VGPR v, lanes 0-15 -> M=v, lanes 16-31 -> M=v+8; N = lane%16
  const int half = lane >> 4;
  #pragma unroll
  for (int v = 0; v < 8; ++v) {
    int row = r0 + v + 8 * half;
    if (row < nrows) C[(size_t)row * Dh + c0 + m] = acc[v];
  }
}

// ---------------------------------------------------------------------------
// Message aggregation: out[dst] += xw[src] * (dinv[src]*dinv[dst]) over edges
// plus self-loops out[n] += xw[n]*dinv[n]^2. 16 threads per item, float4 each.
// ---------------------------------------------------------------------------
__global__ void aggregate_kernel(const float* __restrict__ xw,
                                 const int* __restrict__ src,
                                 const int* __restrict__ dst,
                                 const float* __restrict__ dinv,
                                 float* __restrict__ out, int e, int n) {
  int t = blockIdx.x * blockDim.x + threadIdx.x;
  int total = (e + n) * 16;
  if (t >= total) return;
  int item  = t >> 4;
  int chunk = (t & 15) * 4;
  int s, d;
  if (item < e) { s = src[item]; d = dst[item]; }
  else          { s = item - e;  d = s;         }
  float norm = dinv[s] * dinv[d];
  float4 v = *(const float4*)(xw + (size_t)s * Dh + chunk);
  float* o = out + (size_t)d * Dh + chunk;
  unsafeAtomicAdd(o + 0, v.x * norm);
  unsafeAtomicAdd(o + 1, v.y * norm);
  unsafeAtomicAdd(o + 2, v.z * norm);
  unsafeAtomicAdd(o + 3, v.w * norm);
}

// h[n*64+c] = relu(h[n*64+c] + b[c])
__global__ void bias_relu_kernel(float* __restrict__ h,
                                 const float* __restrict__ b, int total) {
  int t = blockIdx.x * blockDim.x + threadIdx.x;
  if (t >= total) return;
  float v = h[t] + b[t & (Dh - 1)];
  h[t] = v > 0.0f ? v : 0.0f;
}

// global mean pool accumulation: pooled[batch[n]] += h[n]; cnt[batch[n]] += 1
__global__ void pool_kernel(const float* __restrict__ h,
                            const int* __restrict__ batch,
                            float* __restrict__ pooled,
                            float* __restrict__ cnt, int n) {
  int t = blockIdx.x * blockDim.x + threadIdx.x;
  if (t >= n * 16) return;
  int item  = t >> 4;
  int chunk = (t & 15) * 4;
  int g = batch[item];
  float4 v = *(const float4*)(h + (size_t)item * Dh + chunk);
  float* o = pooled + (size_t)g * Dh + chunk;
  unsafeAtomicAdd(o + 0, v.x);
  unsafeAtomicAdd(o + 1, v.y);
  unsafeAtomicAdd(o + 2, v.z);
  unsafeAtomicAdd(o + 3, v.w);
  if (chunk == 0) unsafeAtomicAdd(&cnt[g], 1.0f);
}

// head: logits = (pooled/cnt) @ Wfc + bfc; out = log_softmax(logits)
__global__ void head_kernel(const float* __restrict__ pooled,
                            const float* __restrict__ cnt,
                            const float* __restrict__ Wfc,
                            const float* __restrict__ bfc,
                            float* __restrict__ out, int g_count) {
  int g = blockIdx.x * blockDim.x + threadIdx.x;
  if (g >= g_count) return;
  float inv = 1.0f / fmaxf(cnt[g], 1.0f);
  float p[Dh];
  #pragma unroll
  for (int i = 0; i < Dh; ++i) p[i] = pooled[(size_t)g * Dh + i] * inv;
  float logit[Oc];
  #pragma unroll
  for (int o = 0; o < Oc; ++o) {
    float s = bfc[o];
    #pragma unroll
    for (int i = 0; i < Dh; ++i) s += p[i] * Wfc[i * Oc + o];
    logit[o] = s;
  }
  float mx = logit[0];
  #pragma unroll
  for (int o = 1; o < Oc; ++o) mx = fmaxf(mx, logit[o]);
  float se = 0.0f;
  #pragma unroll
  for (int o = 0; o < Oc; ++o) se += expf(logit[o] - mx);
  float lse = logf(se) + mx;
  #pragma unroll
  for (int o = 0; o < Oc; ++o) out[(size_t)g * Oc + o] = logit[o] - lse;
}

// ---------------------------------------------------------------------------
extern "C" void kernel_launch(void* const* d_in, const int* in_sizes, int n_in,
                              void* d_out, int out_size, void* d_ws, size_t ws_size,
                              hipStream_t stream) {
  const float* x     = (const float*)d_in[0];
  const int*   ei    = (const int*)d_in[1];   // [2,E]: row0 = src, row1 = dst
  const int*   batch = (const int*)d_in[2];
  const float* W1    = (const float*)d_in[3];
  const float* b1    = (const float*)d_in[4];
  const float* W2    = (const float*)d_in[5];
  const float* b2    = (const float*)d_in[6];
  const float* Wfc   = (const float*)d_in[7];
  const float* bfc   = (const float*)d_in[8];
  float* out = (float*)d_out;

  const int N = in_sizes[2];
  const int E = in_sizes[1] / 2;
  const int G = out_size / Oc;
  const int* src = ei;
  const int* dst = ei + E;

  // workspace carve-up (all fp32)
  float* xw     = (float*)d_ws;               // N*64
  float* h      = xw + (size_t)N * Dh;        // N*64
  float* dinv   = h + (size_t)N * Dh;         // N
  float* pooled = dinv + N;                   // G*64
  float* cnt    = pooled + (size_t)G * Dh;    // G

  auto cdiv = [](int a, int b) { return (a + b - 1) / b; };
  const int TPB = 256;
  const int NH  = N * Dh;

  // --- degrees (shared by both layers): deg = 1 + indeg(dst); dinv = rsqrt ---
  fill_kernel<<<cdiv(N, TPB), TPB, 0, stream>>>(dinv, 1.0f, N);
  deg_count_kernel<<<cdiv(E, TPB), TPB, 0, stream>>>(dst, dinv, E);
  rsqrt_kernel<<<cdiv(N, TPB), TPB, 0, stream>>>(dinv, N);

  // --- layer 1: xw = x@W1 ; h = scatter(norm * xw) ; h = relu(h + b1) ---
  gemm64_wmma_kernel<<<cdiv(N, 16), 128, 0, stream>>>(x, W1, xw, N);
  fill_kernel<<<cdiv(NH, TPB), TPB, 0, stream>>>(h, 0.0f, NH);
  aggregate_kernel<<<cdiv((E + N) * 16, TPB), TPB, 0, stream>>>(xw, src, dst, dinv, h, E, N);
  bias_relu_kernel<<<cdiv(NH, TPB), TPB, 0, stream>>>(h, b1, NH);

  // --- layer 2: xw = h@W2 ; h = scatter(norm * xw) ; h = relu(h + b2) ---
  gemm64_wmma_kernel<<<cdiv(N, 16), 128, 0, stream>>>(h, W2, xw, N);
  fill_kernel<<<cdiv(NH, TPB), TPB, 0, stream>>>(h, 0.0f, NH);
  aggregate_kernel<<<cdiv((E + N) * 16, TPB), TPB, 0, stream>>>(xw, src, dst, dinv, h, E, N);
  bias_relu_kernel<<<cdiv(NH, TPB), TPB, 0, stream>>>(h, b2, NH);

  // --- pool + head ---
  fill_kernel<<<cdiv(G * Dh, TPB), TPB, 0, stream>>>(pooled, 0.0f, G * Dh);
  fill_kernel<<<cdiv(G, TPB), TPB, 0, stream>>>(cnt, 0.0f, G);
  pool_kernel<<<cdiv(N * 16, TPB), TPB, 0, stream>>>(h, batch, pooled, cnt, N);
  head_kernel<<<cdiv(G, 128), 128, 0, stream>>>(pooled, cnt, Wfc, bfc, out, G);
}